// Actor_28552942584470
// MI455X (gfx1250) — compile-verified
//
#include <hip/hip_runtime.h>
#include <math.h>

// CDNA5 fp32 WMMA: D(16x16,f32) = A(16x4,f32) * B(4x16,f32) + C
typedef __attribute__((ext_vector_type(2))) float v2f;
typedef __attribute__((ext_vector_type(8))) float v8f;

#define WMMA_F32(a, b, c) \
  __builtin_amdgcn_wmma_f32_16x16x4_f32(false, (a), false, (b), (short)0, (c), false, false)

__device__ __forceinline__ float fast_sigmoid(float z) {
  return __builtin_amdgcn_rcpf(1.0f + __expf(-z));   // v_rcp_f32, ~1 ulp
}
__device__ __forceinline__ float fast_tanh(float z) {
#if __has_builtin(__builtin_amdgcn_tanhf)
  return __builtin_amdgcn_tanhf(z);                  // gfx1250 v_tanh_f32
#else
  const float e = __expf(2.0f * z);
  return 1.0f - 2.0f * __builtin_amdgcn_rcpf(e + 1.0f);
#endif
}

// ---------------- packed-weight layout (floats) -------------------------
// For weight W (K x N): tiles t over N (16 cols), k-steps s (4 rows).
// Lane l (n = t*16 + (l&15), hi = l>>4) needs rows k0 = 4s+2hi, k0+1.
// Packed: P[((t*S + s)*32 + l)*2 + {0,1}]  -> one global_load_b64 per step.
constexpr int S_WC  = 25,  T_WC  = 4;   // K=100, N=50 ->64
constexpr int S_WA0 = 13,  T_WA0 = 16;  // K=50 ->52, N=256
constexpr int S_WA1 = 64,  T_WA1 = 25;  // K=256, N=400
constexpr int S_WP0 = 100, T_WP0 = 16;  // K=400, N=256
constexpr int S_WP1 = 64,  T_WP1 = 16;  // K=256, N=256
constexpr int S_WP2 = 64,  T_WP2 = 1;   // K=256, N=8 ->16

constexpr size_t OFF_WC  = 0;
constexpr size_t OFF_WA0 = OFF_WC  + (size_t)T_WC  * S_WC  * 64;
constexpr size_t OFF_WA1 = OFF_WA0 + (size_t)T_WA0 * S_WA0 * 64;
constexpr size_t OFF_WP0 = OFF_WA1 + (size_t)T_WA1 * S_WA1 * 64;
constexpr size_t OFF_WP1 = OFF_WP0 + (size_t)T_WP0 * S_WP0 * 64;
constexpr size_t OFF_WP2 = OFF_WP1 + (size_t)T_WP1 * S_WP1 * 64;
constexpr size_t PACK_FLOATS = OFF_WP2 + (size_t)T_WP2 * S_WP2 * 64;   // 294144

__device__ __forceinline__ void pack_one(const float* __restrict__ W, float* __restrict__ P,
                                         int K, int N, int S, int idx) {
  const int t   = idx / (S * 32);
  const int r   = idx - t * S * 32;
  const int s   = r >> 5;
  const int l   = r & 31;
  const int m16 = l & 15, hi = l >> 4;
  const int k0  = 4 * s + 2 * hi;
  const int n   = t * 16 + m16;
  const float x = (k0     < K && n < N) ? W[(size_t)k0 * N + n]       : 0.0f;
  const float y = (k0 + 1 < K && n < N) ? W[(size_t)(k0 + 1) * N + n] : 0.0f;
  P[(size_t)idx * 2]     = x;
  P[(size_t)idx * 2 + 1] = y;
}

__global__ void pack_weights(const float* __restrict__ Wc,  const float* __restrict__ Wa0,
                             const float* __restrict__ Wa1, const float* __restrict__ Wp0,
                             const float* __restrict__ Wp1, const float* __restrict__ Wp2,
                             float* __restrict__ P) {
  const int tid    = blockIdx.x * blockDim.x + threadIdx.x;
  const int stride = gridDim.x * blockDim.x;
  for (int i = tid; i < T_WC  * S_WC  * 32; i += stride) pack_one(Wc,  P + OFF_WC,  100,  50, S_WC,  i);
  for (int i = tid; i < T_WA0 * S_WA0 * 32; i += stride) pack_one(Wa0, P + OFF_WA0,  50, 256, S_WA0, i);
  for (int i = tid; i < T_WA1 * S_WA1 * 32; i += stride) pack_one(Wa1, P + OFF_WA1, 256, 400, S_WA1, i);
  for (int i = tid; i < T_WP0 * S_WP0 * 32; i += stride) pack_one(Wp0, P + OFF_WP0, 400, 256, S_WP0, i);
  for (int i = tid; i < T_WP1 * S_WP1 * 32; i += stride) pack_one(Wp1, P + OFF_WP1, 256, 256, S_WP1, i);
  for (int i = tid; i < T_WP2 * S_WP2 * 32; i += stride) pack_one(Wp2, P + OFF_WP2, 256,   8, S_WP2, i);
}

// ---------------- fused actor network -----------------------------------
// Dynamic LDS, float offsets (27264 floats = 109056 B):
//   s_att : [0,     832)    16 x 52
//   s_x   : [832,  4160)    64 x 52   (4 M-tiles of x)
//   s_h   : [4160, 20800)   64 x 260  (4 M-tiles of h; aliased by s_p/s_q later)
//   s_p   : [4160,  8320)   16 x 260
//   s_q   : [8320, 12480)   16 x 260
//   s_pi  : [20800,27264)   16 x 404  (aliases s_g = 16 x 100 in stage 0)
constexpr size_t LDS_FLOATS = 27264;

template<bool PACKED>
__global__ __launch_bounds__(256, 1)
void actor_fused(const float* __restrict__ o,   const float* __restrict__ g,
                 const float* __restrict__ Wc,  const float* __restrict__ bc,
                 const float* __restrict__ Wa0, const float* __restrict__ ba0,
                 const float* __restrict__ Wa1, const float* __restrict__ ba1,
                 const float* __restrict__ Wp0, const float* __restrict__ bp0,
                 const float* __restrict__ Wp1, const float* __restrict__ bp1,
                 const float* __restrict__ Wp2, const float* __restrict__ bp2,
                 const float* __restrict__ P,   float* __restrict__ out)
{
  extern __shared__ float sm[];
  float* s_att = sm;             // 16 x 52 (50 valid)
  float* s_x   = sm + 832;       // 64 x 52
  float* s_h   = sm + 4160;      // 64 x 260 (256 valid)
  float* s_p   = sm + 4160;      // 16 x 260
  float* s_q   = sm + 8320;      // 16 x 260
  float* s_pi  = sm + 20800;     // 16 x 404 (400 valid)
  float* s_g   = sm + 20800;     // 16 x 100

  const int tid  = threadIdx.x;
  const int wave = tid >> 5;
  const int lane = tid & 31;
  const int m16  = lane & 15;
  const int hi   = lane >> 4;
  const int nb0  = blockIdx.x * 16;

  // ---- stage g into LDS ----
  for (int i = tid; i < 16 * 100; i += 256) {
    int r = i / 100, c = i % 100;
    s_g[r * 100 + c] = g[(size_t)(nb0 + r) * 100 + c];
  }
  __syncthreads();

  // ---- stage 0: attention = sigmoid(g @ Wc + bc)  (16 x 50), K=100 ----
  if (wave < 4) {
    const int n = wave * 16 + m16;            // valid < 50
    const v2f* A  = (const v2f*)(s_g + m16 * 100);
    const v2f* B0 = nullptr;
    if constexpr (PACKED) B0 = (const v2f*)(P + OFF_WC) + (size_t)wave * S_WC * 32 + lane;
    const int nc = (n < 50) ? n : 49;
    v8f c = {};
    #pragma unroll 5
    for (int s = 0; s < S_WC; ++s) {
      v2f a = A[2 * s + hi];
      v2f b;
      if constexpr (PACKED) {
        b = B0[s * 32];
      } else {
        const int k0 = 4 * s + 2 * hi;
        float w0 = Wc[(size_t)k0 * 50 + nc];
        float w1 = Wc[(size_t)(k0 + 1) * 50 + nc];
        b.x = (n < 50) ? w0 : 0.0f;
        b.y = (n < 50) ? w1 : 0.0f;
      }
      c = WMMA_F32(a, b, c);
    }
    if (n < 50) {
      const float bias = bc[n];
      #pragma unroll
      for (int v = 0; v < 8; ++v)
        s_att[(v + 8 * hi) * 52 + n] = fast_sigmoid(c[v] + bias);
    }
  }
  __syncthreads();

  // ---- encoder: 2 groups x 4 M-tiles (each M-tile = 2 batch x 8 objects) ----
  for (int grp = 0; grp < 2; ++grp) {
    // build 4 x tiles (64 x 52, zero-padded K)
    for (int i = tid; i < 64 * 52; i += 256) {
      const int rr = i / 52, k = i - rr * 52;
      const int u = rr >> 4, r = rr & 15;
      const int nl  = 8 * grp + 2 * u + (r >> 3);
      const int obj = r & 7;
      float val = 0.0f;
      if (k < 50) {
        int col;
        if      (k < 10) col = k;
        else if (k < 20) col = 130 + (k - 10);
        else if (k < 35) col = 10  + obj * 15 + (k - 20);
        else             col = 140 + obj * 15 + (k - 35);
        val = o[(size_t)(nb0 + nl) * 260 + col] * s_att[nl * 52 + k];
      }
      s_x[rr * 52 + k] = val;
    }
    __syncthreads();

    // stage 1: h = relu(x @ Wa0 + ba0), 4 M-tiles x 2 N-tiles per wave
    {
      const int t0 = 2 * wave;
      const int n0 = t0 * 16 + m16, n1 = n0 + 16;
      const v2f *B0 = nullptr, *B1 = nullptr;
      if constexpr (PACKED) {
        B0 = (const v2f*)(P + OFF_WA0) + (size_t)t0 * S_WA0 * 32 + lane;
        B1 = B0 + (size_t)S_WA0 * 32;
      }
      v8f c0[4] = {}, c1[4] = {};
      for (int s = 0; s < S_WA0; ++s) {
        v2f b0, b1;
        if constexpr (PACKED) {
          b0 = B0[s * 32];
          b1 = B1[s * 32];
        } else {
          const int k0 = 4 * s + 2 * hi;
          const int kc = (k0 < 48) ? k0 : 48;
          float w00 = Wa0[(size_t)kc * 256 + n0], w01 = Wa0[(size_t)(kc + 1) * 256 + n0];
          float w10 = Wa0[(size_t)kc * 256 + n1], w11 = Wa0[(size_t)(kc + 1) * 256 + n1];
          b0.x = (k0     < 50) ? w00 : 0.0f;
          b0.y = (k0 + 1 < 50) ? w01 : 0.0f;
          b1.x = (k0     < 50) ? w10 : 0.0f;
          b1.y = (k0 + 1 < 50) ? w11 : 0.0f;
        }
        #pragma unroll
        for (int u = 0; u < 4; ++u) {
          v2f a = ((const v2f*)(s_x + (u * 16 + m16) * 52))[2 * s + hi];
          c0[u] = WMMA_F32(a, b0, c0[u]);
          c1[u] = WMMA_F32(a, b1, c1[u]);
        }
      }
      const float bias0 = ba0[n0], bias1 = ba0[n1];
      #pragma unroll
      for (int u = 0; u < 4; ++u) {
        #pragma unroll
        for (int v = 0; v < 8; ++v) {
          s_h[(u * 16 + v + 8 * hi) * 260 + n0] = fmaxf(c0[u][v] + bias0, 0.0f);
          s_h[(u * 16 + v + 8 * hi) * 260 + n1] = fmaxf(c1[u][v] + bias1, 0.0f);
        }
      }
    }
    __syncthreads();

    // stage 2 (dominant): y = relu(h @ Wa1 + ba1), object-sum -> pi.
    // Triple N-tile x quad M-tile: 12 wmma per (3 glb + 4 ds) b64 loads.
    {
      const int n0 = wave * 16 + m16, n1 = n0 + 128, n2 = n0 + 256;
      const v2f *B0 = nullptr, *B1 = nullptr, *B2 = nullptr;
      if constexpr (PACKED) {
        B0 = (const v2f*)(P + OFF_WA1) + (size_t)wave * S_WA1 * 32 + lane;
        B1 = B0 + (size_t)8  * S_WA1 * 32;
        B2 = B0 + (size_t)16 * S_WA1 * 32;
      }
      v8f c0[4] = {}, c1[4] = {}, c2[4] = {};
      #pragma unroll 2
      for (int s = 0; s < S_WA1; ++s) {
        v2f b0, b1, b2;
        if constexpr (PACKED) {
          b0 = B0[s * 32];
          b1 = B1[s * 32];
          b2 = B2[s * 32];
        } else {
          const int k0 = 4 * s + 2 * hi;
          b0.x = Wa1[(size_t)k0 * 400 + n0]; b0.y = Wa1[(size_t)(k0 + 1) * 400 + n0];
          b1.x = Wa1[(size_t)k0 * 400 + n1]; b1.y = Wa1[(size_t)(k0 + 1) * 400 + n1];
          b2.x = Wa1[(size_t)k0 * 400 + n2]; b2.y = Wa1[(size_t)(k0 + 1) * 400 + n2];
        }
        #pragma unroll
        for (int u = 0; u < 4; ++u) {
          v2f a = ((const v2f*)(s_h + (u * 16 + m16) * 260))[2 * s + hi];
          c0[u] = WMMA_F32(a, b0, c0[u]);
          c1[u] = WMMA_F32(a, b1, c1[u]);
          c2[u] = WMMA_F32(a, b2, c2[u]);
        }
      }
      const float bias0 = ba1[n0], bias1 = ba1[n1], bias2 = ba1[n2];
      #pragma unroll
      for (int u = 0; u < 4; ++u) {
        float r0 = 0.0f, r1 = 0.0f, r2 = 0.0f;
        #pragma unroll
        for (int v = 0; v < 8; ++v) {
          r0 += fmaxf(c0[u][v] + bias0, 0.0f);
          r1 += fmaxf(c1[u][v] + bias1, 0.0f);
          r2 += fmaxf(c2[u][v] + bias2, 0.0f);
        }
        const int row = (8 * grp + 2 * u + hi) * 404;
        s_pi[row + n0] = r0;
        s_pi[row + n1] = r1;
        s_pi[row + n2] = r2;
      }
    }
    if (wave == 0) {                       // last N-tile t = 24
      const int n = 384 + m16;
      const v2f* B0 = nullptr;
      if constexpr (PACKED) B0 = (const v2f*)(P + OFF_WA1) + (size_t)24 * S_WA1 * 32 + lane;
      v8f c[4] = {};
      #pragma unroll 4
      for (int s = 0; s < S_WA1; ++s) {
        v2f b;
        if constexpr (PACKED) {
          b = B0[s * 32];
        } else {
          const int k0 = 4 * s + 2 * hi;
          b.x = Wa1[(size_t)k0 * 400 + n]; b.y = Wa1[(size_t)(k0 + 1) * 400 + n];
        }
        #pragma unroll
        for (int u = 0; u < 4; ++u) {
          v2f a = ((const v2f*)(s_h + (u * 16 + m16) * 260))[2 * s + hi];
          c[u] = WMMA_F32(a, b, c[u]);
        }
      }
      const float bias = ba1[n];
      #pragma unroll
      for (int u = 0; u < 4; ++u) {
        float r = 0.0f;
        #pragma unroll
        for (int v = 0; v < 8; ++v) r += fmaxf(c[u][v] + bias, 0.0f);
        s_pi[(8 * grp + 2 * u + hi) * 404 + n] = r;
      }
    }
    __syncthreads();
  }

  // ---- stage 3: p = relu(pi @ Wp0 + bp0)  (16 x 256), K=400, 2 tiles/wave ----
  {
    const int t0 = 2 * wave;
    const int n0 = t0 * 16 + m16, n1 = n0 + 16;
    const v2f* A  = (const v2f*)(s_pi + m16 * 404);
    const v2f *B0 = nullptr, *B1 = nullptr;
    if constexpr (PACKED) {
      B0 = (const v2f*)(P + OFF_WP0) + (size_t)t0 * S_WP0 * 32 + lane;
      B1 = B0 + (size_t)S_WP0 * 32;
    }
    v8f c0 = {}, c1 = {};
    #pragma unroll 5
    for (int s = 0; s < S_WP0; ++s) {
      v2f a = A[2 * s + hi];
      v2f b0, b1;
      if constexpr (PACKED) {
        b0 = B0[s * 32];
        b1 = B1[s * 32];
      } else {
        const int k0 = 4 * s + 2 * hi;
        b0.x = Wp0[(size_t)k0 * 256 + n0]; b0.y = Wp0[(size_t)(k0 + 1) * 256 + n0];
        b1.x = Wp0[(size_t)k0 * 256 + n1]; b1.y = Wp0[(size_t)(k0 + 1) * 256 + n1];
      }
      c0 = WMMA_F32(a, b0, c0);
      c1 = WMMA_F32(a, b1, c1);
    }
    const float bias0 = bp0[n0], bias1 = bp0[n1];
    #pragma unroll
    for (int v = 0; v < 8; ++v) {
      s_p[(v + 8 * hi) * 260 + n0] = fmaxf(c0[v] + bias0, 0.0f);
      s_p[(v + 8 * hi) * 260 + n1] = fmaxf(c1[v] + bias1, 0.0f);
    }
  }
  __syncthreads();

  // ---- stage 4: q = relu(p @ Wp1 + bp1)  (16 x 256), K=256, 2 tiles/wave ----
  {
    const int t0 = 2 * wave;
    const int n0 = t0 * 16 + m16, n1 = n0 + 16;
    const v2f* A  = (const v2f*)(s_p + m16 * 260);
    const v2f *B0 = nullptr, *B1 = nullptr;
    if constexpr (PACKED) {
      B0 = (const v2f*)(P + OFF_WP1) + (size_t)t0 * S_WP1 * 32 + lane;
      B1 = B0 + (size_t)S_WP1 * 32;
    }
    v8f c0 = {}, c1 = {};
    #pragma unroll 4
    for (int s = 0; s < S_WP1; ++s) {
      v2f a = A[2 * s + hi];
      v2f b0, b1;
      if constexpr (PACKED) {
        b0 = B0[s * 32];
        b1 = B1[s * 32];
      } else {
        const int k0 = 4 * s + 2 * hi;
        b0.x = Wp1[(size_t)k0 * 256 + n0]; b0.y = Wp1[(size_t)(k0 + 1) * 256 + n0];
        b1.x = Wp1[(size_t)k0 * 256 + n1]; b1.y = Wp1[(size_t)(k0 + 1) * 256 + n1];
      }
      c0 = WMMA_F32(a, b0, c0);
      c1 = WMMA_F32(a, b1, c1);
    }
    const float bias0 = bp1[n0], bias1 = bp1[n1];
    #pragma unroll
    for (int v = 0; v < 8; ++v) {
      s_q[(v + 8 * hi) * 260 + n0] = fmaxf(c0[v] + bias0, 0.0f);
      s_q[(v + 8 * hi) * 260 + n1] = fmaxf(c1[v] + bias1, 0.0f);
    }
  }
  __syncthreads();

  // ---- stage 5: out = tanh(q @ Wp2 + bp2)  (16 x 8), K=256, wave 0 ----
  if (wave == 0) {
    const int n  = m16;                 // valid < 8
    const int nc = n & 7;
    const v2f* A  = (const v2f*)(s_q + m16 * 260);
    const v2f* B0 = nullptr;
    if constexpr (PACKED) B0 = (const v2f*)(P + OFF_WP2) + lane;
    v8f c = {};
    #pragma unroll 4
    for (int s = 0; s < S_WP2; ++s) {
      v2f a = A[2 * s + hi];
      v2f b;
      if constexpr (PACKED) {
        b = B0[s * 32];
      } else {
        const int k0 = 4 * s + 2 * hi;
        float w0 = Wp2[(size_t)k0 * 8 + nc];
        float w1 = Wp2[(size_t)(k0 + 1) * 8 + nc];
        b.x = (n < 8) ? w0 : 0.0f;
        b.y = (n < 8) ? w1 : 0.0f;
      }
      c = WMMA_F32(a, b, c);
    }
    if (n < 8) {
      const float bias = bp2[n];
      #pragma unroll
      for (int v = 0; v < 8; ++v)
        out[(size_t)(nb0 + v + 8 * hi) * 8 + n] = fast_tanh(c[v] + bias);
    }
  }
}

extern "C" void kernel_launch(void* const* d_in, const int* in_sizes, int n_in,
                              void* d_out, int out_size, void* d_ws, size_t ws_size,
                              hipStream_t stream) {
  const float* o   = (const float*)d_in[0];
  const float* g   = (const float*)d_in[1];
  const float* Wc  = (const float*)d_in[2];
  const float* bc  = (const float*)d_in[3];
  const float* Wa0 = (const float*)d_in[4];
  const float* ba0 = (const float*)d_in[5];
  const float* Wa1 = (const float*)d_in[6];
  const float* ba1 = (const float*)d_in[7];
  const float* Wp0 = (const float*)d_in[8];
  const float* bp0 = (const float*)d_in[9];
  const float* Wp1 = (const float*)d_in[10];
  const float* bp1 = (const float*)d_in[11];
  const float* Wp2 = (const float*)d_in[12];
  const float* bp2 = (const float*)d_in[13];

  const int Bn = in_sizes[0] / 260;      // 65536
  dim3 grid(Bn / 16), block(256);
  const size_t lds_bytes = LDS_FLOATS * sizeof(float);   // 109056

  if (ws_size >= PACK_FLOATS * sizeof(float)) {
    float* P = (float*)d_ws;
    pack_weights<<<dim3(288), dim3(256), 0, stream>>>(Wc, Wa0, Wa1, Wp0, Wp1, Wp2, P);
    actor_fused<true><<<grid, block, lds_bytes, stream>>>(o, g, Wc, bc, Wa0, ba0, Wa1, ba1,
                                                          Wp0, bp0, Wp1, bp1, Wp2, bp2,
                                                          P, (float*)d_out);
  } else {
    actor_fused<false><<<grid, block, lds_bytes, stream>>>(o, g, Wc, bc, Wa0, ba0, Wa1, ba1,
                                                           Wp0, bp0, Wp1, bp1, Wp2, bp2,
                                                           nullptr, (float*)d_out);
  }
}